// MainModelEncoder_39178691674343
// MI455X (gfx1250) — compile-verified
//
#include <hip/hip_runtime.h>
#include <hip/hip_bf16.h>
#include <math.h>

typedef __attribute__((ext_vector_type(2))) float v2f;
typedef __attribute__((ext_vector_type(8))) float v8f;

#define HID 128
#define NHEAD 4
#define CH 32

static __device__ __forceinline__ void atomicMaxF(float* addr, float val) {
    int* ai = (int*)addr;
    int old = __float_as_int(*addr);
    while (__int_as_float(old) < val) {
        int assumed = old;
        old = atomicCAS(ai, assumed, __float_as_int(val));
        if (old == assumed) break;
    }
}

// ---------------------------------------------------------------- fill
__global__ void fill_f32(float* __restrict__ p, float v, int n) {
    int i = blockIdx.x * blockDim.x + threadIdx.x;
    if (i < n) p[i] = v;
}

// ---------------------------------------------------------------- embed+concat
__global__ void embed_concat(const int* __restrict__ xn,
                             const float* __restrict__ emb,
                             const float* __restrict__ demb,
                             float* __restrict__ X, int N) {
    int i = blockIdx.x * blockDim.x + threadIdx.x;
    if (i >= N * 64) return;
    int n = i >> 6, c = i & 63;
    float v;
    if (c < 32) {
        v = emb[xn[n * 2] * 32 + c];
    } else {
        int d = xn[n * 2 + 1];
        d = d < 0 ? 0 : (d > 19 ? 19 : d);
        v = demb[d * 32 + (c - 32)];
    }
    X[(size_t)n * 64 + c] = v;
}

// ---------------------------------------------------------------- f32 WMMA GEMM
// C[M x 128] = A[M x K] @ W[K x 128] (+ bias).  One wave -> 16 rows x 128 cols.
// Uses V_WMMA_F32_16X16X4_F32 (full-precision f32 matrix pipe, CDNA5).
__global__ void __launch_bounds__(256) gemm_f32_wmma(
    const float* __restrict__ A, const float* __restrict__ W,
    const float* __restrict__ bias, float* __restrict__ C,
    int M, int K)
{
    __shared__ float sW[HID * HID];            // up to 128x128 weights (64 KB LDS)
    const int tid = threadIdx.x;
    const int tot = K * HID;
    for (int i = tid; i < tot; i += blockDim.x) sW[i] = W[i];
    __syncthreads();

    const int wave = (blockIdx.x * blockDim.x + tid) >> 5;
    const int lane = tid & 31;
    const int m0 = wave << 4;
    if (m0 >= M) return;                       // wave-uniform exit: EXEC stays all-1s

    const int lr = lane & 15;                  // row-in-tile (A) / column (B,C)
    const int lh = lane >> 4;                  // half selector (K pairs / row+8)

    v8f acc[8];
    #pragma unroll
    for (int t = 0; t < 8; ++t) {
        float bv = bias ? bias[t * 16 + lr] : 0.0f;
        #pragma unroll
        for (int r = 0; r < 8; ++r) acc[t][r] = bv;
    }

    int arow = m0 + lr; if (arow >= M) arow = M - 1;   // clamped load, no EXEC change
    const float* ap = A + (size_t)arow * K + (lh << 1);

    for (int kb = 0; kb < K; kb += 4) {
        v2f a = *(const v2f*)(ap + kb);        // A frag: lane = m + (k/2)*16, vgpr = k%2
        const int krow = kb + (lh << 1);
        #pragma unroll
        for (int t = 0; t < 8; ++t) {
            const int col = t * 16 + lr;
            v2f b;                             // B frag: vgpr v holds row krow+v, col lr
            b.x = sW[krow * HID + col];
            b.y = sW[(krow + 1) * HID + col];
            acc[t] = __builtin_amdgcn_wmma_f32_16x16x4_f32(
                false, a, false, b, (short)0, acc[t], false, false);
        }
    }

    const int rbase = m0 + (lh << 3);          // lanes 16..31 hold rows +8
    #pragma unroll
    for (int t = 0; t < 8; ++t) {
        const int col = t * 16 + lr;
        #pragma unroll
        for (int r = 0; r < 8; ++r) {
            int row = rbase + r;
            if (row < M) C[(size_t)row * HID + col] = acc[t][r];
        }
    }
}

// ---------------------------------------------------------------- GAT attention coefs
__global__ void gat_coef(const float* __restrict__ hw,
                         const float* __restrict__ wsrc,
                         const float* __restrict__ wdst,
                         float* __restrict__ asrc, float* __restrict__ adst, int N) {
    int i = blockIdx.x * blockDim.x + threadIdx.x;
    if (i >= N * NHEAD) return;
    int n = i >> 2, hd = i & 3;
    const float* row = hw + (size_t)n * HID + hd * CH;
    float s1 = 0.f, s2 = 0.f;
    #pragma unroll
    for (int c = 0; c < CH; ++c) {
        float v = row[c];
        s1 += v * wsrc[hd * CH + c];
        s2 += v * wdst[hd * CH + c];
    }
    asrc[i] = s1;
    adst[i] = s2;
}

// ---------------------------------------------------------------- GAT softmax pass 1 (max)
__global__ void gat_max(const int* __restrict__ ei, int E, int N,
                        const float* __restrict__ asrc, const float* __restrict__ adst,
                        float* __restrict__ m) {
    int i = blockIdx.x * blockDim.x + threadIdx.x;
    int EL = E + N;
    if (i >= EL * NHEAD) return;
    int e = i >> 2, hd = i & 3;
    int s, d;
    if (e < E) { s = ei[e]; d = ei[E + e]; } else { s = d = e - E; }
    float sc = asrc[s * 4 + hd] + adst[d * 4 + hd];
    sc = sc >= 0.f ? sc : 0.2f * sc;
    atomicMaxF(&m[d * 4 + hd], sc);
}

// ---------------------------------------------------------------- GAT softmax pass 2 (exp+sum)
__global__ void gat_exp(const int* __restrict__ ei, int E, int N,
                        const float* __restrict__ asrc, const float* __restrict__ adst,
                        const float* __restrict__ m,
                        float* __restrict__ es, float* __restrict__ ssum) {
    int i = blockIdx.x * blockDim.x + threadIdx.x;
    int EL = E + N;
    if (i >= EL * NHEAD) return;
    int e = i >> 2, hd = i & 3;
    int s, d;
    if (e < E) { s = ei[e]; d = ei[E + e]; } else { s = d = e - E; }
    float sc = asrc[s * 4 + hd] + adst[d * 4 + hd];
    sc = sc >= 0.f ? sc : 0.2f * sc;
    float mm = m[d * 4 + hd];
    if (!isfinite(mm)) mm = 0.f;
    float ev = __expf(sc - mm);
    es[i] = ev;
    atomicAdd(&ssum[d * 4 + hd], ev);
}

// ---------------------------------------------------------------- GAT aggregate (wave/edge)
__global__ void gat_agg(const int* __restrict__ ei, int E, int N,
                        const float* __restrict__ hw,
                        const float* __restrict__ es, const float* __restrict__ ssum,
                        float* __restrict__ out) {
    int gid = blockIdx.x * blockDim.x + threadIdx.x;
    int e = gid >> 5, lane = gid & 31;
    int EL = E + N;
    if (e >= EL) return;
    int s, d;
    if (e < E) { s = ei[e]; d = ei[E + e]; } else { s = d = e - E; }
    #pragma unroll
    for (int j = 0; j < NHEAD; ++j) {
        float alpha = es[e * 4 + j] / (ssum[d * 4 + j] + 1e-16f);
        int c = j * CH + lane;
        atomicAdd(&out[(size_t)d * HID + c], hw[(size_t)s * HID + c] * alpha);
    }
}

// ---------------------------------------------------------------- h += elu(out + bias)
__global__ void gat_update(float* __restrict__ h, const float* __restrict__ out,
                           const float* __restrict__ bias, int N) {
    int i = blockIdx.x * blockDim.x + threadIdx.x;
    if (i >= N * HID) return;
    float x = out[i] + bias[i & (HID - 1)];
    x = x > 0.f ? x : __expf(x) - 1.f;
    h[i] += x;
}

// ---------------------------------------------------------------- transformer scores
__global__ void tr_score(const int* __restrict__ ei, int E,
                         const float* __restrict__ q, const float* __restrict__ k,
                         float* __restrict__ sc, float* __restrict__ m) {
    int i = blockIdx.x * blockDim.x + threadIdx.x;
    if (i >= E * NHEAD) return;
    int e = i >> 2, hd = i & 3;
    int s = ei[e], d = ei[E + e];
    const float* qp = q + (size_t)d * HID + hd * CH;
    const float* kp = k + (size_t)s * HID + hd * CH;
    float a = 0.f;
    #pragma unroll
    for (int c = 0; c < CH; ++c) a += qp[c] * kp[c];
    a *= 0.17677669529663687f;   // 1/sqrt(32)
    sc[i] = a;
    atomicMaxF(&m[d * 4 + hd], a);
}

__global__ void tr_exp(const int* __restrict__ ei, int E,
                       float* __restrict__ sc, const float* __restrict__ m,
                       float* __restrict__ ssum) {
    int i = blockIdx.x * blockDim.x + threadIdx.x;
    if (i >= E * NHEAD) return;
    int e = i >> 2, hd = i & 3;
    int d = ei[E + e];
    float mm = m[d * 4 + hd];
    if (!isfinite(mm)) mm = 0.f;
    float ev = __expf(sc[i] - mm);
    sc[i] = ev;
    atomicAdd(&ssum[d * 4 + hd], ev);
}

__global__ void tr_agg(const int* __restrict__ ei, int E,
                       const float* __restrict__ v,
                       const float* __restrict__ es, const float* __restrict__ ssum,
                       float* __restrict__ agg) {
    int gid = blockIdx.x * blockDim.x + threadIdx.x;
    int e = gid >> 5, lane = gid & 31;
    if (e >= E) return;
    int s = ei[e], d = ei[E + e];
    #pragma unroll
    for (int j = 0; j < NHEAD; ++j) {
        float alpha = es[e * 4 + j] / (ssum[d * 4 + j] + 1e-16f);
        int c = j * CH + lane;
        atomicAdd(&agg[(size_t)d * HID + c], v[(size_t)s * HID + c] * alpha);
    }
}

__global__ void tr_update(float* __restrict__ h, const float* __restrict__ agg, int N) {
    int i = blockIdx.x * blockDim.x + threadIdx.x;
    if (i >= N * HID) return;
    float x = agg[i];
    x = x > 0.f ? x : __expf(x) - 1.f;
    h[i] += x;
}

// ---------------------------------------------------------------- layernorm (wave/node, wave32)
__global__ void layernorm(float* __restrict__ h, const float* __restrict__ g,
                          const float* __restrict__ b, int N) {
    int gid = blockIdx.x * blockDim.x + threadIdx.x;
    int n = gid >> 5, lane = gid & 31;
    if (n >= N) return;
    float x[4]; float s = 0.f;
    #pragma unroll
    for (int j = 0; j < 4; ++j) { x[j] = h[(size_t)n * HID + j * 32 + lane]; s += x[j]; }
    #pragma unroll
    for (int o = 16; o > 0; o >>= 1) s += __shfl_xor(s, o, 32);
    float mu = s * (1.f / 128.f);
    float vs = 0.f;
    #pragma unroll
    for (int j = 0; j < 4; ++j) { float d = x[j] - mu; vs += d * d; }
    #pragma unroll
    for (int o = 16; o > 0; o >>= 1) vs += __shfl_xor(vs, o, 32);
    float inv = rsqrtf(vs * (1.f / 128.f) + 1e-5f);
    #pragma unroll
    for (int j = 0; j < 4; ++j) {
        int c = j * 32 + lane;
        h[(size_t)n * HID + c] = (x[j] - mu) * inv * g[c] + b[c];
    }
}

// ---------------------------------------------------------------- pooling
__global__ void pool_cnt(const int* __restrict__ batch, float* __restrict__ cnt, int N) {
    int i = blockIdx.x * blockDim.x + threadIdx.x;
    if (i < N) atomicAdd(&cnt[batch[i]], 1.f);
}
__global__ void pool_sum(const float* __restrict__ h, const int* __restrict__ batch,
                         float* __restrict__ sums, int N) {
    int i = blockIdx.x * blockDim.x + threadIdx.x;
    if (i >= N * HID) return;
    int n = i >> 7, c = i & (HID - 1);
    atomicAdd(&sums[(size_t)batch[n] * HID + c], h[i]);
}
__global__ void pool_out(const float* __restrict__ sums, const float* __restrict__ cnt,
                         float* __restrict__ out, int G) {
    int i = blockIdx.x * blockDim.x + threadIdx.x;
    if (i >= G * HID) return;
    int g = i >> 7;
    float c = cnt[g];
    out[i] = sums[i] / (c > 1.f ? c : 1.f);
}

// ================================================================ host
static inline int cdiv(long long a, long long b) { return (int)((a + b - 1) / b); }

extern "C" void kernel_launch(void* const* d_in, const int* in_sizes, int n_in,
                              void* d_out, int out_size, void* d_ws, size_t ws_size,
                              hipStream_t stream) {
    (void)n_in; (void)ws_size;
    const int* xn     = (const int*)d_in[0];
    const int* ei     = (const int*)d_in[1];
    const int* batch  = (const int*)d_in[2];
    const float* emb  = (const float*)d_in[4];
    const float* demb = (const float*)d_in[5];
    const float* pW   = (const float*)d_in[6];
    const float* pb   = (const float*)d_in[7];
    const float* gW   = (const float*)d_in[8];
    const float* gAs  = (const float*)d_in[9];
    const float* gAd  = (const float*)d_in[10];
    const float* gB   = (const float*)d_in[11];
    const float* Wq   = (const float*)d_in[12];
    const float* bq   = (const float*)d_in[13];
    const float* Wk   = (const float*)d_in[14];
    const float* bk   = (const float*)d_in[15];
    const float* Wv   = (const float*)d_in[16];
    const float* bv   = (const float*)d_in[17];
    const float* Wsk  = (const float*)d_in[18];
    const float* bsk  = (const float*)d_in[19];
    const float* lng  = (const float*)d_in[20];
    const float* lnb  = (const float*)d_in[21];
    float* out = (float*)d_out;

    const int N = in_sizes[0] / 2;
    const int E = in_sizes[1] / 2;
    const int G = out_size / HID;
    const int Kproj = in_sizes[6] / HID;   // 64
    const int EL = E + N;

    // workspace layout (floats)
    float* ws = (float*)d_ws;
    size_t off = 0;
    float* X    = ws + off; off += (size_t)N * 64;
    float* h    = ws + off; off += (size_t)N * HID;
    float* hw   = ws + off; off += (size_t)N * HID;   // also q
    float* kb_  = ws + off; off += (size_t)N * HID;   // k
    float* vb_  = ws + off; off += (size_t)N * HID;   // v
    float* agg  = ws + off; off += (size_t)N * HID;   // GAT out / trans agg
    float* asrc = ws + off; off += (size_t)N * NHEAD;
    float* adst = ws + off; off += (size_t)N * NHEAD;
    float* mbuf = ws + off; off += (size_t)N * NHEAD;
    float* sbuf = ws + off; off += (size_t)N * NHEAD;
    float* es   = ws + off; off += (size_t)EL * NHEAD;
    float* sums = ws + off; off += (size_t)G * HID;
    float* cnt  = ws + off; off += (size_t)G;

    const int B = 256;
    const int gemmGrid = cdiv((long long)cdiv(N, 16), 8);   // 8 waves/block

    // 1) embedding + concat, projection GEMM (WMMA f32)
    embed_concat<<<cdiv((long long)N * 64, B), B, 0, stream>>>(xn, emb, demb, X, N);
    gemm_f32_wmma<<<gemmGrid, B, 0, stream>>>(X, pW, pb, h, N, Kproj);

    // 2) three GAT layers
    for (int l = 0; l < 3; ++l) {
        gemm_f32_wmma<<<gemmGrid, B, 0, stream>>>(h, gW + (size_t)l * HID * HID, nullptr, hw, N, HID);
        gat_coef<<<cdiv((long long)N * NHEAD, B), B, 0, stream>>>(
            hw, gAs + l * NHEAD * CH, gAd + l * NHEAD * CH, asrc, adst, N);
        fill_f32<<<cdiv((long long)N * NHEAD, B), B, 0, stream>>>(mbuf, -INFINITY, N * NHEAD);
        fill_f32<<<cdiv((long long)N * NHEAD, B), B, 0, stream>>>(sbuf, 0.f, N * NHEAD);
        fill_f32<<<cdiv((long long)N * HID, B), B, 0, stream>>>(agg, 0.f, N * HID);
        gat_max<<<cdiv((long long)EL * NHEAD, B), B, 0, stream>>>(ei, E, N, asrc, adst, mbuf);
        gat_exp<<<cdiv((long long)EL * NHEAD, B), B, 0, stream>>>(ei, E, N, asrc, adst, mbuf, es, sbuf);
        gat_agg<<<cdiv((long long)EL * 32, B), B, 0, stream>>>(ei, E, N, hw, es, sbuf, agg);
        gat_update<<<cdiv((long long)N * HID, B), B, 0, stream>>>(h, agg, gB + l * HID, N);
    }

    // 3) edge transformer: q,k,v,skip GEMMs
    gemm_f32_wmma<<<gemmGrid, B, 0, stream>>>(h, Wq, bq, hw, N, HID);
    gemm_f32_wmma<<<gemmGrid, B, 0, stream>>>(h, Wk, bk, kb_, N, HID);
    gemm_f32_wmma<<<gemmGrid, B, 0, stream>>>(h, Wv, bv, vb_, N, HID);
    gemm_f32_wmma<<<gemmGrid, B, 0, stream>>>(h, Wsk, bsk, agg, N, HID);   // skip pre-load

    fill_f32<<<cdiv((long long)N * NHEAD, B), B, 0, stream>>>(mbuf, -INFINITY, N * NHEAD);
    fill_f32<<<cdiv((long long)N * NHEAD, B), B, 0, stream>>>(sbuf, 0.f, N * NHEAD);
    tr_score<<<cdiv((long long)E * NHEAD, B), B, 0, stream>>>(ei, E, hw, kb_, es, mbuf);
    tr_exp<<<cdiv((long long)E * NHEAD, B), B, 0, stream>>>(ei, E, es, mbuf, sbuf);
    tr_agg<<<cdiv((long long)E * 32, B), B, 0, stream>>>(ei, E, vb_, es, sbuf, agg);
    tr_update<<<cdiv((long long)N * HID, B), B, 0, stream>>>(h, agg, N);

    // 4) layernorm + mean pooling
    layernorm<<<cdiv((long long)N * 32, B), B, 0, stream>>>(h, lng, lnb, N);
    fill_f32<<<cdiv((long long)G * HID, B), B, 0, stream>>>(sums, 0.f, G * HID);
    fill_f32<<<cdiv((long long)G, B), B, 0, stream>>>(cnt, 0.f, G);
    pool_cnt<<<cdiv((long long)N, B), B, 0, stream>>>(batch, cnt, N);
    pool_sum<<<cdiv((long long)N * HID, B), B, 0, stream>>>(h, batch, sums, N);
    pool_out<<<cdiv((long long)G * HID, B), B, 0, stream>>>(sums, cnt, out, G);
}